// VectorQuantization_5274219839571
// MI455X (gfx1250) — compile-verified
//
#include <hip/hip_runtime.h>

typedef float v2f __attribute__((ext_vector_type(2)));
typedef float v8f __attribute__((ext_vector_type(8)));

namespace {
constexpr int    kB        = 8;
constexpr int    kD        = 128;
constexpr int    kT        = 8192;
constexpr int    kK        = 1024;
constexpr int    kNRows    = kB * kT;                  // 65536 rows of x
constexpr size_t kQElems   = (size_t)kB * kD * kT;     // 8388608
constexpr size_t kIndOff   = kQElems;                  // indices (as float) after quantize
constexpr size_t kLossOff  = kQElems + (size_t)kNRows; // scalar loss last
constexpr int    kWaves    = 8;                        // 256 threads = 8 wave32
constexpr int    kRowsPerBlock = kWaves * 16;          // 128 rows / block
constexpr int    kNBlocks  = kNRows / kRowsPerBlock;   // 512 blocks
constexpr float  kInvTotal = 1.0f / (float)kQElems;    // COMMITMENT_WEIGHT = 1.0
} // namespace

// One wave handles 16 rows of x against the full 1024-entry codebook using
// V_WMMA_F32_16X16X4_F32 (fp32 matrix core path on gfx1250).
__global__ __launch_bounds__(256)
void vq_fused_kernel(const float* __restrict__ x,      // [B, D, T]
                     const float* __restrict__ embed,  // [K, D]
                     float* __restrict__ out,          // quantize | ind | loss
                     float* __restrict__ ws) {         // per-block loss partials
  __shared__ int   s_idx[kWaves * 16];
  __shared__ float s_part[kWaves];

  const int tid  = threadIdx.x;
  const int lane = tid & 31;
  const int wave = tid >> 5;
  const int m    = lane & 15;   // row (A) / code (B,C) sub-index
  const int h    = lane >> 4;   // half-wave select

  const int rowBase = blockIdx.x * kRowsPerBlock + wave * 16;
  const int b  = rowBase >> 13;        // rowBase / kT   (kT = 8192)
  const int t0 = rowBase & (kT - 1);   // rowBase % kT   (tile never crosses b)

  const float* xBase = x + (size_t)b * kD * kT;

  // ---- Preload A tile (16 rows x 128 dims) in WMMA f32 16x16x4 layout ----
  // step s: lanes 0-15 hold A[m][4s+0],A[m][4s+1]; lanes 16-31 hold K=2,3.
  // x is [B,D,T] so a fixed d, consecutive t => coalesced across lanes 0-15.
  float a0[32], a1[32];
  {
    const float* xr = xBase + (t0 + m);
    #pragma unroll
    for (int s = 0; s < 32; ++s) {
      const int d = 4 * s + 2 * h;
      a0[s] = xr[(size_t)d * kT];
      a1[s] = xr[(size_t)(d + 1) * kT];
    }
  }

  float best[8];
  int   bidx[8];
  #pragma unroll
  for (int v = 0; v < 8; ++v) { best[v] = -3.402823466e38f; bidx[v] = 0; }

  // ---- Sweep the codebook in 16-code tiles ----
  for (int kb = 0; kb < kK; kb += 16) {
    const float* er = embed + (size_t)(kb + m) * kD + 2 * h;
    if (kb + 16 < kK)
      __builtin_prefetch(er + 16 * kD, 0, 0);   // next tile -> global_prefetch_b8

    v8f   c   = {0.f, 0.f, 0.f, 0.f, 0.f, 0.f, 0.f, 0.f};
    float e2p = 0.f;
    #pragma unroll
    for (int s = 0; s < 32; ++s) {
      // B layout (4x16): lanes 0-15 K={0,1}, lanes 16-31 K={2,3}; col = m.
      // embed[kb+m][4s+2h], [4s+2h+1] are contiguous -> b64 load.
      const float b0 = er[4 * s];
      const float b1 = er[4 * s + 1];
      e2p += b0 * b0 + b1 * b1;                 // partial ||e||^2 for code kb+m
      v2f av = {a0[s], a1[s]};
      v2f bv = {b0, b1};
      c = __builtin_amdgcn_wmma_f32_16x16x4_f32(false, av, false, bv,
                                                (short)0, c, false, false);
    }
    // lane l and l^16 hold complementary halves of ||e_{kb+m}||^2
    const float e2   = e2p + __shfl_xor(e2p, 16, 32);
    const int   code = kb + m;
    // C layout: lane holds code N = m, rows M = v + 8*h for VGPR v.
    #pragma unroll
    for (int v = 0; v < 8; ++v) {
      const float sc = 2.0f * c[v] - e2;        // -||x||^2 is argmax-invariant
      if (sc > best[v]) { best[v] = sc; bidx[v] = code; }
    }
  }

  // ---- Argmax across the 16 lanes of each half-wave (xor stays in-half) ----
  #pragma unroll
  for (int v = 0; v < 8; ++v) {
    float bv = best[v];
    int   bi = bidx[v];
    #pragma unroll
    for (int off = 1; off < 16; off <<= 1) {
      const float ov = __shfl_xor(bv, off, 32);
      const int   oi = __shfl_xor(bi, off, 32);
      if (ov > bv || (ov == bv && oi < bi)) { bv = ov; bi = oi; }  // first-max tie-break
    }
    if (m == 0) {
      const int r = v + 8 * h;                  // row within this wave's tile
      s_idx[wave * 16 + r] = bi;
      out[kIndOff + (size_t)(rowBase + r)] = (float)bi;
    }
  }
  __syncthreads();

  // ---- Gather codewords, straight-through output, commit-loss partial ----
  // lane covers row r = m, dims d == h (mod 2); halves together cover all 128.
  {
    const int    r   = m;
    const int    rid = s_idx[wave * 16 + r];
    const int    t   = t0 + r;
    const float* er  = embed + (size_t)rid * kD;
    const float* xc  = xBase + t;
    float*       oc  = out + (size_t)b * kD * kT + t;
    float lsum = 0.f;
    #pragma unroll 8
    for (int db = 0; db < 64; ++db) {
      const int   d  = 2 * db + h;
      const float q  = er[d];
      const float xv = xc[(size_t)d * kT];
      const float df = q - xv;                  // quantize - x
      oc[(size_t)d * kT] = xv + df;             // x + stop_grad(quantize - x)
      lsum += df * df;
    }
    #pragma unroll
    for (int off = 1; off < 32; off <<= 1)
      lsum += __shfl_xor(lsum, off, 32);
    if (lane == 0) s_part[wave] = lsum;
  }
  __syncthreads();
  if (tid == 0) {
    float bs = 0.f;
    #pragma unroll
    for (int w = 0; w < kWaves; ++w) bs += s_part[w];
    ws[blockIdx.x] = bs;                        // deterministic per-block partial
  }
}

// Fixed-order reduction of the 512 block partials -> deterministic scalar loss.
__global__ void vq_loss_reduce_kernel(const float* __restrict__ ws,
                                      float* __restrict__ out) {
  if (threadIdx.x == 0 && blockIdx.x == 0) {
    float s = 0.f;
    for (int i = 0; i < kNBlocks; ++i) s += ws[i];
    out[kLossOff] = s * kInvTotal;              // COMMITMENT_WEIGHT == 1.0
  }
}

extern "C" void kernel_launch(void* const* d_in, const int* in_sizes, int n_in,
                              void* d_out, int out_size, void* d_ws, size_t ws_size,
                              hipStream_t stream) {
  (void)in_sizes; (void)n_in; (void)out_size; (void)ws_size;
  const float* x     = (const float*)d_in[0];   // [8, 128, 8192]
  const float* embed = (const float*)d_in[1];   // [1024, 128]
  float*       out   = (float*)d_out;           // quantize | indices | loss
  float*       ws    = (float*)d_ws;            // >= 512 floats of scratch

  vq_fused_kernel<<<kNBlocks, 256, 0, stream>>>(x, embed, out, ws);
  vq_loss_reduce_kernel<<<1, 32, 0, stream>>>(ws, out);
}